// EENDSSLoss_18519898981067
// MI455X (gfx1250) — compile-verified
//
#include <hip/hip_runtime.h>
#include <hip/hip_bf16.h>
#include <math.h>

#define C_SPK 3
#define B_BATCH 32
#define T_LEN 128000
#define T4 (T_LEN / 4)            // 32000 float4 per channel
#define T_SUB 1000
#define T_F 8000
#define CHUNKS 32
#define F4_PER_CHUNK (T4 / CHUNKS) // 1000
#define EPSF 1e-8f

// ---- d_ws layout (floats) ----
// [WS_GRAM  .. +32*24)  final 21 gram scalars per batch (padded to 24)
//     per batch: [0..8]=cross[i*3+j], [9..11]=sum(sep), [12..14]=sum(src),
//                [15..17]=sum(sep^2), [18..20]=sum(src^2)
// [WS_DIARG .. +32*24)  raw WMMA diar gram G[6][4] per batch
// [WS_PART  .. +32*CHUNKS*24) per-(batch,chunk) partials of the 21 scalars
#define WS_GRAM  0
#define WS_DIARG (B_BATCH * 24)
#define WS_PART  (2 * B_BATCH * 24)

typedef float v2f_t __attribute__((ext_vector_type(2)));
typedef float v8f_t __attribute__((ext_vector_type(8)));

__device__ __forceinline__ float clogf(float x) {
    return fmaxf(logf(x), -100.0f);
}

// ---------------------------------------------------------------------------
// Kernel A: HBM-bound streaming gram reduction (coalesced float4, one pass).
// grid = (CHUNKS, B); block = 256. Writes 21 partials per block, no atomics.
// ---------------------------------------------------------------------------
__global__ __launch_bounds__(256) void eendss_gram_partial(
    const float* __restrict__ sep, const float* __restrict__ src,
    float* __restrict__ ws)
{
    const int chunk = blockIdx.x;
    const int b     = blockIdx.y;
    const int tid   = threadIdx.x;

    const float4* S4 = (const float4*)sep;
    const float4* X4 = (const float4*)src;
    const int base0 = (b * 3 + 0) * T4 + chunk * F4_PER_CHUNK;
    const int base1 = (b * 3 + 1) * T4 + chunk * F4_PER_CHUNK;
    const int base2 = (b * 3 + 2) * T4 + chunk * F4_PER_CHUNK;

    float c00 = 0.f, c01 = 0.f, c02 = 0.f;
    float c10 = 0.f, c11 = 0.f, c12 = 0.f;
    float c20 = 0.f, c21 = 0.f, c22 = 0.f;
    float ss0 = 0.f, ss1 = 0.f, ss2 = 0.f;   // sum sep
    float ts0 = 0.f, ts1 = 0.f, ts2 = 0.f;   // sum src
    float q0 = 0.f, q1 = 0.f, q2 = 0.f;      // sum sep^2
    float r0 = 0.f, r1 = 0.f, r2 = 0.f;      // sum src^2

    for (int i = tid; i < F4_PER_CHUNK; i += 256) {
        if (i + 256 < F4_PER_CHUNK) {
            __builtin_prefetch(&S4[base0 + i + 256], 0, 1);  // global_prefetch_b8
            __builtin_prefetch(&X4[base0 + i + 256], 0, 1);
        }
        float4 s0 = S4[base0 + i];
        float4 s1 = S4[base1 + i];
        float4 s2 = S4[base2 + i];
        float4 x0 = X4[base0 + i];
        float4 x1 = X4[base1 + i];
        float4 x2 = X4[base2 + i];

#define ACC(comp)                                                           \
        {                                                                   \
            float a0 = s0.comp, a1 = s1.comp, a2 = s2.comp;                 \
            float b0 = x0.comp, b1 = x1.comp, b2 = x2.comp;                 \
            c00 = fmaf(a0, b0, c00); c01 = fmaf(a0, b1, c01);               \
            c02 = fmaf(a0, b2, c02);                                        \
            c10 = fmaf(a1, b0, c10); c11 = fmaf(a1, b1, c11);               \
            c12 = fmaf(a1, b2, c12);                                        \
            c20 = fmaf(a2, b0, c20); c21 = fmaf(a2, b1, c21);               \
            c22 = fmaf(a2, b2, c22);                                        \
            q0 = fmaf(a0, a0, q0); q1 = fmaf(a1, a1, q1);                   \
            q2 = fmaf(a2, a2, q2);                                          \
            r0 = fmaf(b0, b0, r0); r1 = fmaf(b1, b1, r1);                   \
            r2 = fmaf(b2, b2, r2);                                          \
            ss0 += a0; ss1 += a1; ss2 += a2;                                \
            ts0 += b0; ts1 += b1; ts2 += b2;                                \
        }
        ACC(x) ACC(y) ACC(z) ACC(w)
#undef ACC
    }

    float acc[21] = { c00, c01, c02, c10, c11, c12, c20, c21, c22,
                      ss0, ss1, ss2, ts0, ts1, ts2, q0, q1, q2, r0, r1, r2 };

    // wave32 reduction, then cross-wave via LDS (deterministic tree order)
#pragma unroll
    for (int k = 0; k < 21; ++k) {
        float v = acc[k];
#pragma unroll
        for (int off = 16; off > 0; off >>= 1)
            v += __shfl_xor(v, off, 32);
        acc[k] = v;
    }

    __shared__ float sm[8][21];
    const int wv = tid >> 5, ln = tid & 31;
    if (ln == 0) {
#pragma unroll
        for (int k = 0; k < 21; ++k) sm[wv][k] = acc[k];
    }
    __syncthreads();
    if (tid < 21) {
        float s = 0.f;
#pragma unroll
        for (int w = 0; w < 8; ++w) s += sm[w][tid];
        ws[WS_PART + (b * CHUNKS + chunk) * 24 + tid] = s;
    }
}

// ---------------------------------------------------------------------------
// Kernel A2: fold chunk partials in fixed order. grid = B, block = 32.
// ---------------------------------------------------------------------------
__global__ __launch_bounds__(32) void eendss_gram_reduce(float* __restrict__ ws)
{
    const int b = blockIdx.x;
    const int k = threadIdx.x;
    if (k < 21) {
        float s = 0.f;
        for (int c = 0; c < CHUNKS; ++c)
            s += ws[WS_PART + (b * CHUNKS + c) * 24 + k];
        ws[WS_GRAM + b * 24 + k] = s;
    }
}

// ---------------------------------------------------------------------------
// Kernel B: diarization contraction via V_WMMA_F32_16X16X4_F32.
//   A rows 0-2 : (log p - log(1-p))[t, i]      (clamped logs)
//   A rows 3-5 : log(1-p)[t, i]
//   B cols 0-2 : y[t, j] = labels[b, 8t, j]
//   B col    3 : 1.0
// => G[i][j] = Σ (lp-l1p)·y ;  G[3+i][3] = Σ l1p
//
// Branchless inner loop: per-lane selection folded into loop-invariant
// weights; all loads unconditional with lane-clamped (in-bounds) addresses,
// so there is zero exec-mask churn around the WMMA (EXEC stays all-1s).
// One wave per batch, K=4 timesteps per WMMA, 250 WMMAs.
// ---------------------------------------------------------------------------
__global__ __launch_bounds__(32) void eendss_diar_wmma(
    const float* __restrict__ diar, const float* __restrict__ labels,
    float* __restrict__ ws)
{
    const int b    = blockIdx.x;
    const int lane = threadIdx.x;
    const int mn   = lane & 15;          // row (A) / col (B) index
    const int kb   = (lane >> 4) * 2;    // K base: lanes 0-15 -> K{0,1}, 16-31 -> K{2,3}

    // Loop-invariant per-lane weights (rows/cols >= used range get weight 0).
    const int   chA   = (mn < 3) ? mn : ((mn < 6) ? (mn - 3) : 0); // clamped, in-bounds
    const float w_lp  = (mn < 3) ? 1.0f : 0.0f;                    // + lp   for rows 0-2
    const float w_l1p = (mn < 3) ? -1.0f : ((mn < 6) ? 1.0f : 0.0f); // -l1p rows 0-2, +l1p rows 3-5
    const int   chB   = (mn < 3) ? mn : 0;                         // clamped, in-bounds
    const float w_y   = (mn < 3) ? 1.0f : 0.0f;                    // label weight cols 0-2
    const float w_one = (mn == 3) ? 1.0f : 0.0f;                   // ones column 3

    const float* dbase = diar   + (b * T_SUB) * 3;
    const float* lbase = labels + (b * T_F)   * 3;

    v8f_t cfrag = {};
    for (int t0 = 0; t0 < T_SUB; t0 += 4) {
        v2f_t afrag, bfrag;
#pragma unroll
        for (int kk = 0; kk < 2; ++kk) {
            const int t  = t0 + kb + kk;
            const float p   = dbase[t * 3 + chA];
            const float lp  = clogf(p);
            const float l1p = clogf(1.0f - p);
            const float y   = lbase[(t * 8) * 3 + chB];
            afrag[kk] = fmaf(w_lp, lp, w_l1p * l1p);
            bfrag[kk] = fmaf(w_y, y, w_one);
        }
        // D = A(16x4) * B(4x16) + C, f32 in / f32 accumulate
        cfrag = __builtin_amdgcn_wmma_f32_16x16x4_f32(
            false, afrag, false, bfrag, (short)0, cfrag, false, false);
    }

    // C layout: VGPR v, lanes 0-15 -> (M=v, N=lane). Need M=0..5, N=0..3.
    if (lane < 4) {
#pragma unroll
        for (int v = 0; v < 6; ++v)
            ws[WS_DIARG + b * 24 + v * 4 + lane] = cfrag[v];
    }
}

// ---------------------------------------------------------------------------
// Kernel C: permutation search + existence BCE + final mixing. 1 wave,
// lane = batch, shfl reduction, 5 float outputs.
// ---------------------------------------------------------------------------
__global__ __launch_bounds__(32) void eendss_final(
    const float* __restrict__ exist, const int* __restrict__ nspk,
    const float* __restrict__ ws, float* __restrict__ out)
{
    const int b = threadIdx.x;           // 0..31 == B
    const float Tf = (float)T_LEN;

    const float* g = ws + WS_GRAM + b * 24;
    float esq[3], tsqv[3];
#pragma unroll
    for (int i = 0; i < 3; ++i) {
        esq[i]  = g[15 + i] - g[9 + i]  * g[9 + i]  / Tf;
        tsqv[i] = g[18 + i] - g[12 + i] * g[12 + i] / Tf;
    }
    float S[3][3];
#pragma unroll
    for (int i = 0; i < 3; ++i)
#pragma unroll
        for (int j = 0; j < 3; ++j) {
            const float dotc  = g[i * 3 + j] - g[9 + i] * g[12 + j] / Tf;
            const float alpha = dotc / (tsqv[j] + EPSF);
            const float a2t   = alpha * alpha * tsqv[j];
            const float sig   = a2t + EPSF;
            const float noise = esq[i] - 2.0f * alpha * dotc + a2t + EPSF;
            S[i][j] = 10.0f * log10f(sig / noise);
        }

    const float* gd = ws + WS_DIARG + b * 24;
    float D[3][3];
#pragma unroll
    for (int i = 0; i < 3; ++i)
#pragma unroll
        for (int j = 0; j < 3; ++j)
            D[i][j] = -(gd[i * 4 + j] + gd[(3 + i) * 4 + 3]) * (1.0f / (float)T_SUB);

    const int P[6][3] = {{0,1,2},{0,2,1},{1,0,2},{1,2,0},{2,0,1},{2,1,0}};
    const int ns = nspk[b];
    const int n  = ns < 1 ? 1 : (ns > 3 ? 3 : ns);
    const float nf = (float)n;

    float best_s = INFINITY, best_d = INFINITY;
#pragma unroll
    for (int p = 0; p < 6; ++p) {
        bool valid = true;
        float ssum = 0.f, dsum = 0.f;
#pragma unroll
        for (int k = 0; k < 3; ++k) {
            valid = valid && ((P[p][k] < n) || (k >= n));
            if (k < n) { ssum += S[P[p][k]][k]; dsum += D[P[p][k]][k]; }
        }
        if (valid) {
            best_s = fminf(best_s, -ssum / nf);
            best_d = fminf(best_d,  dsum / nf);
        }
    }

    const int ne = ns < 3 ? ns : 3;
    float esum = 0.f;
#pragma unroll
    for (int c = 0; c < 4; ++c) {
        const float tv = (c < ne) ? 1.0f : 0.0f;
        const float pv = exist[b * 4 + c];
        esum += -(tv * clogf(pv) + (1.0f - tv) * clogf(1.0f - pv));
    }

    float vs = best_s, vd = best_d, ve = esum;
#pragma unroll
    for (int off = 16; off > 0; off >>= 1) {
        vs += __shfl_xor(vs, off, 32);
        vd += __shfl_xor(vd, off, 32);
        ve += __shfl_xor(ve, off, 32);
    }
    if (b == 0) {
        const float loss_sisnr = vs / 32.0f;
        const float loss_diar  = vd / 32.0f;
        const float loss_exist = ve / 128.0f;
        out[0] = loss_sisnr + 0.2f * loss_diar + 0.2f * loss_exist;
        out[1] = loss_sisnr;
        out[2] = loss_diar;
        out[3] = loss_exist;
        out[4] = -loss_sisnr;
    }
}

// ---------------------------------------------------------------------------
extern "C" void kernel_launch(void* const* d_in, const int* in_sizes, int n_in,
                              void* d_out, int out_size, void* d_ws, size_t ws_size,
                              hipStream_t stream) {
    const float* sep    = (const float*)d_in[0];  // separated  (32,3,128000)
    const float* diar   = (const float*)d_in[1];  // diar_probs (32,1000,3)
    const float* exist  = (const float*)d_in[2];  // exist_probs(32,4)
    const float* src    = (const float*)d_in[3];  // sources    (32,3,128000)
    const float* labels = (const float*)d_in[4];  // labels     (32,8000,3)
    const int*   nspk   = (const int*)d_in[5];    // num_speakers (32,)
    float* out = (float*)d_out;
    float* ws  = (float*)d_ws;

    eendss_gram_partial<<<dim3(CHUNKS, B_BATCH), 256, 0, stream>>>(sep, src, ws);
    eendss_gram_reduce <<<dim3(B_BATCH),          32, 0, stream>>>(ws);
    eendss_diar_wmma   <<<dim3(B_BATCH),          32, 0, stream>>>(diar, labels, ws);
    eendss_final       <<<dim3(1),                32, 0, stream>>>(exist, nspk, ws, out);
}